// SimCSEModel_90829968376315
// MI455X (gfx1250) — compile-verified
//
#include <hip/hip_runtime.h>
#include <math.h>

typedef __attribute__((ext_vector_type(2))) float v2f;
typedef __attribute__((ext_vector_type(8))) float v8f;
typedef __attribute__((ext_vector_type(4))) unsigned int u32x4;
typedef __attribute__((ext_vector_type(4))) int i32x4;
typedef __attribute__((ext_vector_type(8))) int i32x8;

#define B_N 8192
#define H_N 768
#define L_N 64
#define P_N 4
#define TEMP_INV 20.0f      // 1 / 0.05
#define EPS_N 1e-8f
#define XPAD 772            // kernel A LDS row stride (768+4), conflict-free

// Kernel B tile: TDM pads 4 dwords after every 256 dwords -> stride 780
#define ROWS_B 32
#define XSTR 780
#define LIDX(m, c) ((m) * XSTR + (c) + 4 * ((c) >> 8))

// Probe the DEVICE pass only: the host (x86) pass never has AMDGPU builtins.
#if defined(__HIP_DEVICE_COMPILE__)
  #if __has_builtin(__builtin_amdgcn_tensor_load_to_lds) && \
      __has_builtin(__builtin_amdgcn_s_wait_tensorcnt)
    #define USE_TDM 1
  #else
    #error "gfx1250 device pass lacks TDM builtins (tensor_load_to_lds / s_wait_tensorcnt)"
  #endif
#endif

#ifdef USE_TDM
// DMA a rows x 768-f32 row-major tile from global into LDS via the Tensor
// Data Mover, inserting a 4-dword pad every 256 dwords (rows = 3x256 dwords,
// so row boundaries align with pad boundaries -> LDS stride 780 dwords,
// 780 % 64 = 12 -> the 16 A-matrix rows land in 16 distinct banks).
__device__ __forceinline__ void tdm_load_tile(const float* gptr,
                                              unsigned lds_off, int rows) {
    const unsigned long long ga = (unsigned long long)(uintptr_t)gptr;
    u32x4 g0;
    g0.x = 1u;                                          // count=1, load, user
    g0.y = lds_off;                                     // lds_addr
    g0.z = (unsigned)(ga & 0xffffffffu);                // global_addr[31:0]
    g0.w = (unsigned)((ga >> 32) & 0x01ffffffu) | (2u << 30);  // [56:32]|type=2
    i32x8 g1;
    g1[0] = (2 << 16)        // data_size = 4 bytes
          | (1 << 20)        // pad_enable
          | (7 << 22)        // pad_interval: 256 dwords
          | (3 << 25);       // pad_amount:   4 dwords
    g1[1] = (H_N & 0xffff) << 16;                       // tensor_dim0[15:0]
    g1[2] = ((H_N >> 16) & 0xffff) | ((rows & 0xffff) << 16); // dim0 hi|dim1 lo
    g1[3] = ((rows >> 16) & 0xffff) | ((H_N & 0xffff) << 16); // dim1 hi|tile_dim0
    g1[4] = rows & 0xffff;                              // tile_dim1
    g1[5] = H_N;                                        // tensor_dim0_stride lo
    g1[6] = 0;
    g1[7] = 0;
    const i32x4 z4 = {0, 0, 0, 0};
#if defined(__clang_major__) && (__clang_major__ >= 23)
    const i32x8 z8 = {0, 0, 0, 0, 0, 0, 0, 0};
    __builtin_amdgcn_tensor_load_to_lds(g0, g1, z4, z4, z8, 0);
#else
    __builtin_amdgcn_tensor_load_to_lds(g0, g1, z4, z4, 0);
#endif
}
#endif  // USE_TDM

// ---------------------------------------------------------------------------
// Kernel A: n = normalize_rows(tanh(x @ W^T + b))  via V_WMMA_F32_16X16X4_F32
// ---------------------------------------------------------------------------
__global__ __launch_bounds__(256)
void proj_norm_kernel(const float* __restrict__ x, const float* __restrict__ W,
                      const float* __restrict__ bias, float* __restrict__ out)
{
    __shared__ float xs[16 * XPAD];
    __shared__ float part[256];
    __shared__ float invn[16];

    const int tid  = threadIdx.x;
    const int lane = tid & 31;
    const int wave = tid >> 5;
    const int row0 = blockIdx.x * 16;

    for (int idx = tid; idx < 16 * (H_N / 4); idx += 256) {
        int m  = idx / (H_N / 4);
        int c4 = (idx % (H_N / 4)) * 4;
        const float4 v = *(const float4*)(x + (size_t)(row0 + m) * H_N + c4);
        float* dst = &xs[m * XPAD + c4];
        dst[0] = v.x; dst[1] = v.y; dst[2] = v.z; dst[3] = v.w;
    }
    __syncthreads();

    const int m     = lane & 15;
    const int khalf = (lane < 16) ? 0 : 2;
    const int rbase = (lane < 16) ? 0 : 8;

    v8f acc[6];
    for (int t = 0; t < 6; ++t) {
        const int n0 = (wave + 8 * t) * 16;
        const float* wrow = W + (size_t)(n0 + m) * H_N;  // B[k][n] = W[n][k]
        v8f c = {0.f, 0.f, 0.f, 0.f, 0.f, 0.f, 0.f, 0.f};
        for (int kk = 0; kk < H_N; kk += 4) {
            v2f a, bf;
            a.x  = xs[m * XPAD + kk + khalf];
            a.y  = xs[m * XPAD + kk + khalf + 1];
            bf.x = wrow[kk + khalf];
            bf.y = wrow[kk + khalf + 1];
            c = __builtin_amdgcn_wmma_f32_16x16x4_f32(false, a, false, bf,
                                                      (short)0, c, false, false);
        }
        acc[t] = c;
    }
    __syncthreads();

    for (int t = 0; t < 6; ++t) {
        const int col = (wave + 8 * t) * 16 + m;
        const float bb = bias[col];
        for (int r = 0; r < 8; ++r)
            xs[(rbase + r) * XPAD + col] = tanhf(acc[t][r] + bb);
    }
    __syncthreads();

    {
        const int row = tid & 15, g = tid >> 4;
        float s = 0.f;
        for (int c = g * 48; c < g * 48 + 48; ++c) {
            const float v = xs[row * XPAD + c];
            s += v * v;
        }
        part[tid] = s;
    }
    __syncthreads();
    if (tid < 16) {
        float s = 0.f;
        for (int g = 0; g < 16; ++g) s += part[g * 16 + tid];
        invn[tid] = 1.0f / fmaxf(sqrtf(s), EPS_N);
    }
    __syncthreads();

    for (int idx = tid; idx < 16 * (H_N / 4); idx += 256) {
        int mm = idx / (H_N / 4);
        int c4 = (idx % (H_N / 4)) * 4;
        const float inv = invn[mm];
        float4 v;
        v.x = xs[mm * XPAD + c4 + 0] * inv;
        v.y = xs[mm * XPAD + c4 + 1] * inv;
        v.z = xs[mm * XPAD + c4 + 2] * inv;
        v.w = xs[mm * XPAD + c4 + 3] * inv;
        *(float4*)(out + (size_t)(row0 + mm) * H_N + c4) = v;
    }
}

// ---------------------------------------------------------------------------
// Kernel B: logZ[i] = logsumexp_j( dot(n1[i],n2[j]) / TEMP ), S never stored.
// 32 anchor rows per block (TDM-staged into LDS); each wave applies one B
// fragment of n2 to two A fragments -> 2 WMMAs per global load pair.
// ---------------------------------------------------------------------------
__global__ __launch_bounds__(256)
void sim_lse_kernel(const float* __restrict__ n1, const float* __restrict__ n2,
                    float* __restrict__ logZ)
{
    __shared__ float xs[ROWS_B * XSTR];  // 99840 B; reused for LSE partials

    const int tid  = threadIdx.x;
    const int lane = tid & 31;
    const int wave = tid >> 5;
    const int row0 = blockIdx.x * ROWS_B;

#ifdef USE_TDM
    if (wave == 0) {
        tdm_load_tile(n1 + (size_t)row0 * H_N, (unsigned)(uintptr_t)&xs[0],
                      ROWS_B);
        __builtin_amdgcn_s_wait_tensorcnt(0);
    }
#else
    // Host-parse / non-TDM fallback: same padded LDS layout.
    for (int idx = tid; idx < ROWS_B * (H_N / 4); idx += 256) {
        int mm = idx / (H_N / 4);
        int c4 = (idx % (H_N / 4)) * 4;
        const float4 v = *(const float4*)(n1 + (size_t)(row0 + mm) * H_N + c4);
        float* dst = &xs[LIDX(mm, c4)];
        dst[0] = v.x; dst[1] = v.y; dst[2] = v.z; dst[3] = v.w;
    }
#endif
    __syncthreads();

    const int m     = lane & 15;
    const int khalf = (lane < 16) ? 0 : 2;

    float mx[16], sm[16];
    for (int q = 0; q < 16; ++q) { mx[q] = -1e30f; sm[q] = 0.f; }

    for (int t = 0; t < B_N / 128; ++t) {      // 64 j-chunks per wave
        const int j0 = (wave + 8 * t) * 16;
        const float* brow = n2 + (size_t)(j0 + m) * H_N;  // B[k][n] = n2[j0+n][k]
        v8f c0 = {0.f, 0.f, 0.f, 0.f, 0.f, 0.f, 0.f, 0.f};
        v8f c1 = {0.f, 0.f, 0.f, 0.f, 0.f, 0.f, 0.f, 0.f};
        for (int kk = 0; kk < H_N; kk += 4) {
            const int ko = kk + khalf + 4 * (kk >> 8);   // padded LDS offset
            v2f a0, a1, bf;
            a0.x = xs[m * XSTR + ko];
            a0.y = xs[m * XSTR + ko + 1];
            a1.x = xs[(m + 16) * XSTR + ko];
            a1.y = xs[(m + 16) * XSTR + ko + 1];
            bf.x = brow[kk + khalf];
            bf.y = brow[kk + khalf + 1];
            c0 = __builtin_amdgcn_wmma_f32_16x16x4_f32(false, a0, false, bf,
                                                       (short)0, c0, false, false);
            c1 = __builtin_amdgcn_wmma_f32_16x16x4_f32(false, a1, false, bf,
                                                       (short)0, c1, false, false);
        }
        for (int r = 0; r < 8; ++r) {          // online logsumexp updates
            float v  = c0[r] * TEMP_INV;
            float nm = fmaxf(mx[r], v);
            sm[r] = sm[r] * __expf(mx[r] - nm) + __expf(v - nm);
            mx[r] = nm;
            v  = c1[r] * TEMP_INV;
            nm = fmaxf(mx[8 + r], v);
            sm[8 + r] = sm[8 + r] * __expf(mx[8 + r] - nm) + __expf(v - nm);
            mx[8 + r] = nm;
        }
    }
    __syncthreads();  // done with n1 tile -> reuse xs for partials

    for (int q = 0; q < 16; ++q) {
        xs[(tid * 16 + q) * 2 + 0] = mx[q];
        xs[(tid * 16 + q) * 2 + 1] = sm[q];
    }
    __syncthreads();

    if (tid < 32) {
        const int half  = tid >> 4;            // 0: rows 0-15, 1: rows 16-31
        const int rowh  = tid & 15;
        const int r     = rowh & 7;
        const int lbase = (rowh < 8) ? 0 : 16; // half-wave holding this row
        const int q     = half * 8 + r;
        float gm = -1e30f;
        for (int w = 0; w < 8; ++w)
            for (int l = 0; l < 16; ++l)
                gm = fmaxf(gm, xs[((w * 32 + lbase + l) * 16 + q) * 2]);
        float tot = 0.f;
        for (int w = 0; w < 8; ++w)
            for (int l = 0; l < 16; ++l) {
                const int src = w * 32 + lbase + l;
                tot += xs[(src * 16 + q) * 2 + 1] *
                       __expf(xs[(src * 16 + q) * 2] - gm);
            }
        logZ[row0 + tid] = gm + __logf(tot);
    }
}

// ---------------------------------------------------------------------------
// Kernel C: per-anchor loss from the P=4 positive pairs (one wave each).
// ---------------------------------------------------------------------------
__global__ __launch_bounds__(128)
void pos_loss_kernel(const float* __restrict__ n1, const float* __restrict__ n2,
                     const int* __restrict__ label, const int* __restrict__ pos,
                     const float* __restrict__ logZ, float* __restrict__ lossi)
{
    __shared__ float terms[4];
    const int i    = blockIdx.x;
    const int lane = threadIdx.x & 31;
    const int wave = threadIdx.x >> 5;
    const int j    = pos[i * P_N + wave];

    const float* a  = n1 + (size_t)i * H_N;
    const float* bb = n2 + (size_t)j * H_N;
    float d = 0.f;
    for (int k = lane; k < H_N; k += 32) d += a[k] * bb[k];

    const int* li = label + i * L_N;
    const int* lj = label + j * L_N;
    float inter = 0.f, si = 0.f, sj = 0.f;
    for (int k = lane; k < L_N; k += 32) {
        const float ai = (float)li[k], aj = (float)lj[k];
        inter += ai * aj; si += ai; sj += aj;
    }
    for (int off = 16; off > 0; off >>= 1) {
        d     += __shfl_xor(d, off, 32);
        inter += __shfl_xor(inter, off, 32);
        si    += __shfl_xor(si, off, 32);
        sj    += __shfl_xor(sj, off, 32);
    }
    if (lane == 0) {
        const float S  = d * TEMP_INV;
        const float un = si + sj - inter;
        const float J  = (un > 0.f) ? inter / fmaxf(un, 1.0f) : 0.f;
        terms[wave] = J * (S - logZ[i]);
    }
    __syncthreads();
    if (threadIdx.x == 0)
        lossi[i] = -(terms[0] + terms[1] + terms[2] + terms[3]) / (float)P_N;
}

// ---------------------------------------------------------------------------
// Kernel D: deterministic final sum.
// ---------------------------------------------------------------------------
__global__ __launch_bounds__(256)
void reduce_kernel(const float* __restrict__ lossi, float* __restrict__ out)
{
    __shared__ float s[256];
    float acc = 0.f;
    for (int i = threadIdx.x; i < B_N; i += 256) acc += lossi[i];
    s[threadIdx.x] = acc;
    __syncthreads();
    for (int st = 128; st > 0; st >>= 1) {
        if (threadIdx.x < st) s[threadIdx.x] += s[threadIdx.x + st];
        __syncthreads();
    }
    if (threadIdx.x == 0) out[0] = s[0];
}

extern "C" void kernel_launch(void* const* d_in, const int* in_sizes, int n_in,
                              void* d_out, int out_size, void* d_ws, size_t ws_size,
                              hipStream_t stream) {
    const float* x1    = (const float*)d_in[0];
    const float* x2    = (const float*)d_in[1];
    const float* W     = (const float*)d_in[2];
    const float* b     = (const float*)d_in[3];
    const int*   label = (const int*)d_in[4];
    const int*   pos   = (const int*)d_in[5];
    float* out = (float*)d_out;

    float* n1    = (float*)d_ws;
    float* n2    = n1 + (size_t)B_N * H_N;
    float* logZ  = n2 + (size_t)B_N * H_N;
    float* lossi = logZ + B_N;

    proj_norm_kernel<<<B_N / 16,     256, 0, stream>>>(x1, W, b, n1);
    proj_norm_kernel<<<B_N / 16,     256, 0, stream>>>(x2, W, b, n2);
    sim_lse_kernel  <<<B_N / ROWS_B, 256, 0, stream>>>(n1, n2, logZ);
    pos_loss_kernel <<<B_N,          128, 0, stream>>>(n1, n2, label, pos, logZ, lossi);
    reduce_kernel   <<<1,            256, 0, stream>>>(lossi, out);
}